// rdf_27968827031656
// MI455X (gfx1250) — compile-verified
//
#include <hip/hip_runtime.h>
#include <math.h>

#define NATOMS   1024
#define NBINS    100
#define NBLOCKS  64
#define IPB      (NATOMS / NBLOCKS)   // 16 i-atoms per block
#define THREADS  256                  // 8 waves (wave32)
#define WAVES    (THREADS / 32)
#define HPAD     112                  // padded per-wave histogram stride

typedef __attribute__((ext_vector_type(4))) unsigned int v4u;
typedef __attribute__((ext_vector_type(8))) int          v8i;
typedef __attribute__((ext_vector_type(4))) int          v4i;
typedef __attribute__((ext_vector_type(2))) float        v2f;
typedef __attribute__((ext_vector_type(8))) float        v8f;

// ---------------------------------------------------------------------------
// TDM prefetch of the whole xyz array (1024*3 f32 = 12KB) into LDS.
// D# per CDNA5 ISA: group0 = {count/lds_addr/global_addr/
// type=2}, group1 = {data_size=4B, tensor_dim0=3072, tile_dim0=3072, stride}.
// clang-23 builtin: 6 args (u32x4, i32x8, i32x4, i32x4, i32x8, i32 cpol).
// ---------------------------------------------------------------------------
__device__ __forceinline__ void tdm_prefetch_xyz(const float* gptr, unsigned lds_off) {
    unsigned long long ga = (unsigned long long)(size_t)gptr;
    v4u g0;
    g0.x = 1u;                                            // count=1, user mode
    g0.y = lds_off;                                       // lds_addr [63:32]
    g0.z = (unsigned)(ga & 0xFFFFFFFFu);                  // global_addr [95:64]
    g0.w = ((unsigned)(ga >> 32) & 0x01FFFFFFu)           // global_addr [120:96]
         | (2u << 30);                                    // type=2 ("image")
    v8i g1;
    g1[0] = (int)(2u << 16);                              // data_size=2 (4 bytes)
    g1[1] = (int)((3072u & 0xFFFFu) << 16);               // tensor_dim0 lo16
    g1[2] = (int)(1u << 16);                              // tensor_dim0 hi=0 | tensor_dim1=1
    g1[3] = (int)(3072u << 16);                           // tensor_dim1 hi=0 | tile_dim0=3072
    g1[4] = 0;                                            // tile_dim1=0, tile_dim2=0
    g1[5] = 3072;                                         // tensor_dim0_stride lo32
    g1[6] = 0;
    g1[7] = 0;
    v4i gz4 = {0, 0, 0, 0};
    v8i gz8 = {0, 0, 0, 0, 0, 0, 0, 0};
    __builtin_amdgcn_tensor_load_to_lds(g0, g1, gz4, gz4, gz8, 0);
}

// ---------------------------------------------------------------------------
// Pair kernel: all-pairs min-image distances + windowed Gaussian smearing
// into per-wave LDS histograms; deterministic per-block partial written to ws.
// ---------------------------------------------------------------------------
__global__ __launch_bounds__(THREADS) void rdf_pair_kernel(
        const float* __restrict__ xyz, const float* __restrict__ cell,
        float* __restrict__ partial) {
    __shared__ float sxyz[3 * NATOMS];
    __shared__ float whist[WAVES * HPAD];

    // CDNA5 TDM async staging (one issue from wave 0; TENSORcnt is per-wave)
    if (threadIdx.x < 32) {
        unsigned lds_off = (unsigned)(unsigned long long)(size_t)(void*)sxyz;
        tdm_prefetch_xyz(xyz, lds_off);
        __builtin_amdgcn_s_wait_tensorcnt(0);
    }
    // Fallback copy (TDM descriptor semantics not hardware-verified; final LDS
    // contents rely on these stores; only wave 0's are issued after the tensor wait).
    for (int k = threadIdx.x; k < 3 * NATOMS; k += THREADS) sxyz[k] = xyz[k];
    for (int k = threadIdx.x; k < WAVES * HPAD; k += THREADS) whist[k] = 0.0f;
    __syncthreads();

    const float Lx = cell[0], Ly = cell[1], Lz = cell[2];
    const float w     = 5.0f / 99.0f;        // Gaussian center spacing == width
    const float invw  = 99.0f / 5.0f;
    const float coeff = -0.5f * invw * invw; // -196.02
    const float cut2  = 30.25f;              // (end + 0.5)^2

    const int i  = blockIdx.x * IPB + (threadIdx.x >> 4);
    const float xi = sxyz[3 * i + 0];
    const float yi = sxyz[3 * i + 1];
    const float zi = sxyz[3 * i + 2];
    float* hist = &whist[(threadIdx.x >> 5) * HPAD];

    for (int j = (threadIdx.x & 15); j < NATOMS; j += 16) {
        float dx = sxyz[3 * j + 0] - xi;
        float dy = sxyz[3 * j + 1] - yi;
        float dz = sxyz[3 * j + 2] - zi;
        // minimum-image wrap, matching reference shift = (-(d>=L/2)+(d<-L/2))*L
        dx += (dx < -0.5f * Lx ? Lx : 0.0f) - (dx >= 0.5f * Lx ? Lx : 0.0f);
        dy += (dy < -0.5f * Ly ? Ly : 0.0f) - (dy >= 0.5f * Ly ? Ly : 0.0f);
        dz += (dz < -0.5f * Lz ? Lz : 0.0f) - (dz >= 0.5f * Lz ? Lz : 0.0f);
        float dsq = dx * dx + dy * dy + dz * dz;
        if (dsq < cut2 && dsq != 0.0f) {
            float dist = sqrtf(dsq);
            float tc = dist * invw;
            int lo = (int)tc - 13; if (lo < 0) lo = 0;     // +/-14 widths:
            int hi = (int)tc + 14; if (hi > 99) hi = 99;   // beyond is fp32 zero
            for (int b = lo; b <= hi; ++b) {
                float dlt = dist - (float)b * w;
                float g = __expf(coeff * dlt * dlt);
                atomicAdd(&hist[b], g);                    // ds_add_f32
            }
        }
    }
    __syncthreads();
    for (int b = threadIdx.x; b < NBINS; b += THREADS) {
        float s = 0.0f;
        for (int wv = 0; wv < WAVES; ++wv) s += whist[wv * HPAD + b];
        partial[blockIdx.x * NBINS + b] = s;
    }
}

// ---------------------------------------------------------------------------
// Finalize: WMMA ones-GEMM reduction of 64x100 partials, normalize, emit
// count (100) | bins (101) | rdf_out (100). Single wave32, EXEC all ones.
// ---------------------------------------------------------------------------
__global__ __launch_bounds__(32) void rdf_finalize_kernel(
        const float* __restrict__ partial, float* __restrict__ out) {
    __shared__ float cnt[NBINS + 12];
    const int lane = threadIdx.x;

    v2f A; A.x = 1.0f; A.y = 1.0f;            // 16x4 all-ones A (both K slots)
    const int nbc  = lane & 15;               // column within bin chunk
    const int krow = (lane < 16) ? 0 : 2;     // B-matrix K rows held by lane

    for (int c = 0; c < 7; ++c) {             // 7 chunks of 16 bins (112 >= 100)
        const int nb = c * 16 + nbc;
        v8f acc = {};
        for (int k = 0; k < 16; ++k) {        // 16 K-steps of 4 blocks = 64
            float b0 = 0.0f, b1 = 0.0f;
            if (nb < NBINS) {
                b0 = partial[(4 * k + krow + 0) * NBINS + nb];
                b1 = partial[(4 * k + krow + 1) * NBINS + nb];
            }
            v2f B; B.x = b0; B.y = b1;
            acc = __builtin_amdgcn_wmma_f32_16x16x4_f32(
                false, A, false, B, (short)0, acc, false, false);
        }
        // every output row equals the column sum; row M=0 lives in acc[0] of lanes 0..15
        if (lane < 16 && (c * 16 + lane) < NBINS) cnt[c * 16 + lane] = acc[0];
    }
    __syncthreads();

    float total = 0.0f;
    for (int b = 0; b < NBINS; ++b) total += cnt[b];
    const float inv_total = 1.0f / total;

    for (int idx = lane; idx < 3 * NBINS + 1; idx += 32) {
        if (idx < NBINS) {
            out[idx] = cnt[idx] * inv_total;                    // count
        } else if (idx < 2 * NBINS + 1) {
            out[idx] = 0.05f * (float)(idx - NBINS);            // bins
        } else {
            int b = idx - (2 * NBINS + 1);                      // rdf_out
            double r0 = 0.05 * (double)b, r1 = 0.05 * (double)(b + 1);
            double ratio = (r1 * r1 * r1 - r0 * r0 * r0) / 125.0;  // vol_bin/V
            out[idx] = (float)((double)(cnt[b] * inv_total) / ratio);
        }
    }
}

extern "C" void kernel_launch(void* const* d_in, const int* in_sizes, int n_in,
                              void* d_out, int out_size, void* d_ws, size_t ws_size,
                              hipStream_t stream) {
    const float* xyz  = (const float*)d_in[0];   // (1024,3) f32
    const float* cell = (const float*)d_in[1];   // (3,) f32
    float* out     = (float*)d_out;              // 301 f32
    float* partial = (float*)d_ws;               // NBLOCKS*NBINS f32 = 25.6KB
    rdf_pair_kernel<<<NBLOCKS, THREADS, 0, stream>>>(xyz, cell, partial);
    rdf_finalize_kernel<<<1, 32, 0, stream>>>(partial, out);
}